// HGT_3461743640867
// MI455X (gfx1250) — compile-verified
//
#include <hip/hip_runtime.h>
#include <math.h>

typedef __attribute__((ext_vector_type(2))) float v2f;
typedef __attribute__((ext_vector_type(8))) float v8f;

#define NNODE 10000
#define HIDC  128
#define INDIMC 901
#define EEDGE 100000
#define NTYPES 8
#define ETYPES 18
#define OUTC  64

__device__ __forceinline__ float sigmoidf_(float x){ return 1.0f/(1.0f+expf(-x)); }

// order-preserving float<->uint mapping for atomic max
__device__ __forceinline__ unsigned ordf(float f){
  unsigned u = __float_as_uint(f);
  return (u & 0x80000000u) ? ~u : (u | 0x80000000u);
}
__device__ __forceinline__ float unordf(unsigned u){
  return (u & 0x80000000u) ? __uint_as_float(u ^ 0x80000000u) : __uint_as_float(~u);
}

// ---------------------------------------------------------------------------
// fp32 WMMA GEMM: C[M,N] = epi(A[M,K] @ W[N,K]^T + bias[N])
// One wave computes a 32x32 tile of C (2x2 sub-tiles of 16x16) with four
// independent V_WMMA_F32_16X16X4_F32 accumulator chains; A/B fragments are
// shared across sub-tiles (2 dwords of memory per WMMA instead of 4).
// Fragment layouts per ISA: A 16x4 lanes=M, K = vgpr + 2*(lane>=16);
// B 4x16 lanes=N, K = vgpr + 2*(lane>=16); C vgpr r -> M = r + 8*(lane>=16).
// M need not be a multiple of 32: the second M sub-tile clamps load addresses
// (per-lane select, EXEC stays all-1s) and skips stores wave-uniformly.
// EPI: 0=none 1=relu 2=sigmoid 3=skip-blend (beta=sigmoid(*skip))
// ---------------------------------------------------------------------------
template<int EPI>
__global__ __launch_bounds__(256) void gemm_wmma(
    const float* __restrict__ A, const float* __restrict__ W,
    const float* __restrict__ bias, float* __restrict__ C,
    int M, int N, int K,
    const float* __restrict__ skip, const float* __restrict__ hold)
{
  const int lane = threadIdx.x & 31;
  const int wave = blockIdx.x * (blockDim.x >> 5) + (threadIdx.x >> 5);
  const int ntn  = N >> 5;                     // 32-wide N tiles
  const int tm   = wave / ntn;
  const int tn   = wave - tm * ntn;
  if (tm * 32 >= M) return;                    // wave-uniform guard
  const int half = lane >> 4;
  const int l16  = lane & 15;
  const int r0 = tm * 32;                      // first 16-row group
  const int r1 = tm * 32 + 16;                 // second 16-row group (may OOB)
  int r1l = r1 + l16; if (r1l > M - 1) r1l = M - 1;   // clamp loads only
  const float* Ap0 = A + (size_t)(r0 + l16) * K;
  const float* Ap1 = A + (size_t)r1l * K;
  const float* Wp0 = W + (size_t)(tn * 32 + l16) * K;
  const float* Wp1 = W + (size_t)(tn * 32 + 16 + l16) * K;
  const int kb = half * 2;                     // K offset for this lane half
  v8f c00 = {}, c01 = {}, c10 = {}, c11 = {};
  int k0 = 0;
  for (; k0 + 4 <= K; k0 += 4) {
    v2f a0, a1, b0, b1;
    a0.x = Ap0[k0 + kb];  a0.y = Ap0[k0 + kb + 1];
    a1.x = Ap1[k0 + kb];  a1.y = Ap1[k0 + kb + 1];
    b0.x = Wp0[k0 + kb];  b0.y = Wp0[k0 + kb + 1];
    b1.x = Wp1[k0 + kb];  b1.y = Wp1[k0 + kb + 1];
    c00 = __builtin_amdgcn_wmma_f32_16x16x4_f32(false, a0, false, b0, (short)0, c00, false, false);
    c01 = __builtin_amdgcn_wmma_f32_16x16x4_f32(false, a0, false, b1, (short)0, c01, false, false);
    c10 = __builtin_amdgcn_wmma_f32_16x16x4_f32(false, a1, false, b0, (short)0, c10, false, false);
    c11 = __builtin_amdgcn_wmma_f32_16x16x4_f32(false, a1, false, b1, (short)0, c11, false, false);
  }
  if (k0 < K) {                                // ragged K tail (K=901)
    const int ka = k0 + kb;
    const bool g0 = ka < K, g1 = ka + 1 < K;
    v2f a0, a1, b0, b1;
    a0.x = g0 ? Ap0[ka] : 0.f;  a0.y = g1 ? Ap0[ka + 1] : 0.f;
    a1.x = g0 ? Ap1[ka] : 0.f;  a1.y = g1 ? Ap1[ka + 1] : 0.f;
    b0.x = g0 ? Wp0[ka] : 0.f;  b0.y = g1 ? Wp0[ka + 1] : 0.f;
    b1.x = g0 ? Wp1[ka] : 0.f;  b1.y = g1 ? Wp1[ka + 1] : 0.f;
    c00 = __builtin_amdgcn_wmma_f32_16x16x4_f32(false, a0, false, b0, (short)0, c00, false, false);
    c01 = __builtin_amdgcn_wmma_f32_16x16x4_f32(false, a0, false, b1, (short)0, c01, false, false);
    c10 = __builtin_amdgcn_wmma_f32_16x16x4_f32(false, a1, false, b0, (short)0, c10, false, false);
    c11 = __builtin_amdgcn_wmma_f32_16x16x4_f32(false, a1, false, b1, (short)0, c11, false, false);
  }
  const int ccol0 = tn * 32 + l16;
  const int ccol1 = ccol0 + 16;
  const float bv0 = bias ? bias[ccol0] : 0.f;
  const float bv1 = bias ? bias[ccol1] : 0.f;
  float beta = 0.f;
  if (EPI == 3) beta = sigmoidf_(skip[0]);
  #pragma unroll
  for (int r = 0; r < 8; ++r) {
    const int row0 = r0 + half * 8 + r;
    float v0 = c00[r] + bv0;
    float v1 = c01[r] + bv1;
    if (EPI == 1) { v0 = fmaxf(v0, 0.f); v1 = fmaxf(v1, 0.f); }
    if (EPI == 2) { v0 = sigmoidf_(v0);  v1 = sigmoidf_(v1); }
    if (EPI == 3) {
      v0 = beta * v0 + (1.f - beta) * hold[(size_t)row0 * N + ccol0];
      v1 = beta * v1 + (1.f - beta) * hold[(size_t)row0 * N + ccol1];
    }
    C[(size_t)row0 * N + ccol0] = v0;
    C[(size_t)row0 * N + ccol1] = v1;
  }
  if (r1 < M) {                                // wave-uniform: tile fully valid
    #pragma unroll
    for (int r = 0; r < 8; ++r) {
      const int row1 = r1 + half * 8 + r;
      float v0 = c10[r] + bv0;
      float v1 = c11[r] + bv1;
      if (EPI == 1) { v0 = fmaxf(v0, 0.f); v1 = fmaxf(v1, 0.f); }
      if (EPI == 2) { v0 = sigmoidf_(v0);  v1 = sigmoidf_(v1); }
      if (EPI == 3) {
        v0 = beta * v0 + (1.f - beta) * hold[(size_t)row1 * N + ccol0];
        v1 = beta * v1 + (1.f - beta) * hold[(size_t)row1 * N + ccol1];
      }
      C[(size_t)row1 * N + ccol0] = v0;
      C[(size_t)row1 * N + ccol1] = v1;
    }
  }
}

// ---------------------------------------------------------------------------
// Per-head relation transform: Y[n, h*16+e] = sum_d X[n, h*16+d] * R[h][d][e]
// einsum('nhd,hde->nhe'); one wave per (16-row tile, head); K=16 -> 4 WMMAs.
// ---------------------------------------------------------------------------
__global__ __launch_bounds__(256) void rel_wmma(
    const float* __restrict__ X, const float* __restrict__ R, float* __restrict__ Y)
{
  const int lane = threadIdx.x & 31;
  const int wave = blockIdx.x * (blockDim.x >> 5) + (threadIdx.x >> 5);
  const int mt = wave >> 3;
  const int h  = wave & 7;
  if (mt * 16 >= NNODE) return;
  const int half = lane >> 4;
  const int l16  = lane & 15;
  const float* Xp = X + (size_t)(mt * 16 + l16) * HIDC + h * 16;
  const float* Rp = R + h * 256;
  const int kb = half * 2;
  v8f acc = {};
  #pragma unroll
  for (int k0 = 0; k0 < 16; k0 += 4) {
    v2f a, b;
    a.x = Xp[k0 + kb];
    a.y = Xp[k0 + kb + 1];
    b.x = Rp[(k0 + kb)     * 16 + l16];
    b.y = Rp[(k0 + kb + 1) * 16 + l16];
    acc = __builtin_amdgcn_wmma_f32_16x16x4_f32(false, a, false, b, (short)0, acc,
                                                false, false);
  }
  const int rbase = mt * 16 + half * 8;
  #pragma unroll
  for (int r = 0; r < 8; ++r)
    Y[(size_t)(rbase + r) * HIDC + h * 16 + l16] = acc[r];
}

// ---------------------------------------------------------------------------
// Edge-phase kernels (segment softmax + aggregation via L2 atomics)
// ---------------------------------------------------------------------------
__global__ void seg_init(unsigned* __restrict__ segmax, float* __restrict__ segsum, int n){
  int i = blockIdx.x * blockDim.x + threadIdx.x;
  if (i < n){ segmax[i] = 0x007fffffu; /* ordf(-inf) */ segsum[i] = 0.f; }
}

__global__ void alpha_max(const float* __restrict__ q, const float* __restrict__ krel,
                          const int* __restrict__ src, const int* __restrict__ dst,
                          const float* __restrict__ prel, float* __restrict__ alpha,
                          unsigned* __restrict__ segmax, int E)
{
  int t = blockIdx.x * blockDim.x + threadIdx.x;
  if (t >= E * 8) return;
  int e = t >> 3, h = t & 7;
  int s = src[e], d = dst[e];
  const float4* qp = (const float4*)(q    + (size_t)d * HIDC + h * 16);
  const float4* kp = (const float4*)(krel + (size_t)s * HIDC + h * 16);
  float a = 0.f;
  #pragma unroll
  for (int i = 0; i < 4; ++i){
    float4 qv = qp[i], kv = kp[i];
    a += qv.x*kv.x + qv.y*kv.y + qv.z*kv.z + qv.w*kv.w;
  }
  a *= prel[h] * 0.25f;                 // 1/sqrt(D), D=16
  alpha[t] = a;
  atomicMax(&segmax[d * 8 + h], ordf(a));
}

__global__ void exp_sum(float* __restrict__ alpha, const int* __restrict__ dst,
                        const unsigned* __restrict__ segmax, float* __restrict__ segsum, int E)
{
  int t = blockIdx.x * blockDim.x + threadIdx.x;
  if (t >= E * 8) return;
  int e = t >> 3, h = t & 7;
  int d = dst[e];
  float ex = expf(alpha[t] - unordf(segmax[d * 8 + h]));
  alpha[t] = ex;
  atomicAdd(&segsum[d * 8 + h], ex);
}

__global__ void msg_scatter(const float* __restrict__ alpha, const float* __restrict__ vrel,
                            const int* __restrict__ src, const int* __restrict__ dst,
                            const float* __restrict__ segsum, float* __restrict__ agg, int E)
{
  int t = blockIdx.x * blockDim.x + threadIdx.x;
  if (t >= E * 8) return;
  int e = t >> 3, h = t & 7;
  int s = src[e], d = dst[e];
  float w = alpha[t] / (segsum[d * 8 + h] + 1e-16f);
  const float4* vp = (const float4*)(vrel + (size_t)s * HIDC + h * 16);
  float* ap = agg + (size_t)d * HIDC + h * 16;
  #pragma unroll
  for (int i = 0; i < 4; ++i){
    float4 vv = vp[i];
    atomicAdd(ap + 4*i + 0, vv.x * w);
    atomicAdd(ap + 4*i + 1, vv.y * w);
    atomicAdd(ap + 4*i + 2, vv.z * w);
    atomicAdd(ap + 4*i + 3, vv.w * w);
  }
}

__global__ void gelu_ip(float* __restrict__ x, unsigned n){
  unsigned i = blockIdx.x * blockDim.x + threadIdx.x;
  if (i < n){
    float v = x[i];
    x[i] = 0.5f * v * (1.f + erff(v * 0.70710678118654752f));
  }
}

// ---------------------------------------------------------------------------
extern "C" void kernel_launch(void* const* d_in, const int* in_sizes, int n_in,
                              void* d_out, int out_size, void* d_ws, size_t ws_size,
                              hipStream_t stream)
{
  (void)in_sizes; (void)n_in; (void)out_size; (void)ws_size;
  // src/dst node-type index for each of the 18 edge types
  static const int esrc[ETYPES] = {7,1,7,1,1,2,2,2,2,5,5,0,5,5,0,0,0,3};
  static const int edst[ETYPES] = {1,7,0,3,6,1,6,3,4,1,0,5,3,4,2,3,4,4};

  // ---- unpack inputs (setup_inputs() insertion order, pytree depth-first) ----
  int idx = 0;
  const float* x[NTYPES];
  for (int i = 0; i < NTYPES; ++i) x[i] = (const float*)d_in[idx++];
  const int* ei[ETYPES];
  for (int j = 0; j < ETYPES; ++j) ei[j] = (const int*)d_in[idx++];
  const float *linW[NTYPES], *linb[NTYPES];
  for (int i = 0; i < NTYPES; ++i){ linW[i]=(const float*)d_in[idx++]; linb[i]=(const float*)d_in[idx++]; }
  struct Lay {
    const float *kW[NTYPES], *kb[NTYPES], *qW[NTYPES], *qb[NTYPES];
    const float *vW[NTYPES], *vb[NTYPES], *aW[NTYPES], *ab[NTYPES];
    const float *skip[NTYPES];
    const float *arel[ETYPES], *mrel[ETYPES], *prel[ETYPES];
  } L[2];
  for (int l = 0; l < 2; ++l){
    for (int i = 0; i < NTYPES; ++i){ L[l].kW[i]=(const float*)d_in[idx++]; L[l].kb[i]=(const float*)d_in[idx++]; }
    for (int i = 0; i < NTYPES; ++i){ L[l].qW[i]=(const float*)d_in[idx++]; L[l].qb[i]=(const float*)d_in[idx++]; }
    for (int i = 0; i < NTYPES; ++i){ L[l].vW[i]=(const float*)d_in[idx++]; L[l].vb[i]=(const float*)d_in[idx++]; }
    for (int i = 0; i < NTYPES; ++i){ L[l].aW[i]=(const float*)d_in[idx++]; L[l].ab[i]=(const float*)d_in[idx++]; }
    for (int i = 0; i < NTYPES; ++i)  L[l].skip[i]=(const float*)d_in[idx++];
    for (int j = 0; j < ETYPES; ++j)  L[l].arel[j]=(const float*)d_in[idx++];
    for (int j = 0; j < ETYPES; ++j)  L[l].mrel[j]=(const float*)d_in[idx++];
    for (int j = 0; j < ETYPES; ++j)  L[l].prel[j]=(const float*)d_in[idx++];
  }
  const float* outW = (const float*)d_in[idx++];
  const float* outb = (const float*)d_in[idx++];

  // ---- workspace carve ----
  const size_t NH = (size_t)NNODE * HIDC;          // 1.28M floats per node type
  float* w = (float*)d_ws;
  float* hA   = w; w += NTYPES * NH;
  float* hB   = w; w += NTYPES * NH;
  float* kbuf = w; w += NTYPES * NH;
  float* qbuf = w; w += NTYPES * NH;
  float* vbuf = w; w += NTYPES * NH;
  float* agg  = w; w += NTYPES * NH;
  float* krel = w; w += NH;
  float* vrel = w; w += NH;
  float* alpha = w; w += (size_t)EEDGE * 8;
  unsigned* segmax = (unsigned*)w; w += (size_t)NNODE * 8;
  float* segsum = w; w += (size_t)NNODE * 8;

  const int E8blk   = (EEDGE * 8 + 255) / 256;     // 3125
  const int SEGblk  = (NNODE * 8 + 255) / 256;     // 313
  const int RELblk  = 625;                         // 625*8 waves = 5000 tiles
  // 32x32 tiles: ceil(M/32) * (N/32) waves, 8 waves per block
  auto gblocks = [](int M, int N){
    int tiles = ((M + 31) / 32) * (N / 32);
    return (tiles + 7) / 8;
  };

  // ---- input projection + ReLU ----
  for (int nt = 0; nt < NTYPES; ++nt)
    gemm_wmma<1><<<gblocks(NNODE, HIDC), 256, 0, stream>>>(
        x[nt], linW[nt], linb[nt], hA + nt*NH, NNODE, HIDC, INDIMC, nullptr, nullptr);

  float* hc = hA;
  float* hn = hB;
  for (int l = 0; l < 2; ++l){
    const Lay& P = L[l];
    // K/Q/V projections
    for (int nt = 0; nt < NTYPES; ++nt){
      gemm_wmma<0><<<gblocks(NNODE, HIDC), 256, 0, stream>>>(
          hc + nt*NH, P.kW[nt], P.kb[nt], kbuf + nt*NH, NNODE, HIDC, HIDC, nullptr, nullptr);
      gemm_wmma<0><<<gblocks(NNODE, HIDC), 256, 0, stream>>>(
          hc + nt*NH, P.qW[nt], P.qb[nt], qbuf + nt*NH, NNODE, HIDC, HIDC, nullptr, nullptr);
      gemm_wmma<0><<<gblocks(NNODE, HIDC), 256, 0, stream>>>(
          hc + nt*NH, P.vW[nt], P.vb[nt], vbuf + nt*NH, NNODE, HIDC, HIDC, nullptr, nullptr);
    }
    hipMemsetAsync(agg, 0, NTYPES * NH * sizeof(float), stream);
    // edge types
    for (int et = 0; et < ETYPES; ++et){
      const int s = esrc[et], d = edst[et];
      const int* srcp = ei[et];
      const int* dstp = ei[et] + EEDGE;
      rel_wmma<<<RELblk, 256, 0, stream>>>(kbuf + s*NH, P.arel[et], krel);
      rel_wmma<<<RELblk, 256, 0, stream>>>(vbuf + s*NH, P.mrel[et], vrel);
      seg_init<<<SEGblk, 256, 0, stream>>>(segmax, segsum, NNODE * 8);
      alpha_max<<<E8blk, 256, 0, stream>>>(qbuf + d*NH, krel, srcp, dstp,
                                           P.prel[et], alpha, segmax, EEDGE);
      exp_sum<<<E8blk, 256, 0, stream>>>(alpha, dstp, segmax, segsum, EEDGE);
      msg_scatter<<<E8blk, 256, 0, stream>>>(alpha, vrel, srcp, dstp, segsum,
                                             agg + d*NH, EEDGE);
    }
    // GELU(agg) then A-projection with learned skip blend
    gelu_ip<<<(unsigned)((NTYPES*NH + 255)/256), 256, 0, stream>>>(agg, (unsigned)(NTYPES*NH));
    for (int nt = 0; nt < NTYPES; ++nt)
      gemm_wmma<3><<<gblocks(NNODE, HIDC), 256, 0, stream>>>(
          agg + nt*NH, P.aW[nt], P.ab[nt], hn + nt*NH, NNODE, HIDC, HIDC,
          P.skip[nt], hc + nt*NH);
    float* t = hc; hc = hn; hn = t;
  }

  // ---- output projection + sigmoid: miRNA (nt 0), disease (nt 1) ----
  float* out = (float*)d_out;
  gemm_wmma<2><<<gblocks(NNODE, OUTC), 256, 0, stream>>>(
      hc + 0*NH, outW, outb, out,                       NNODE, OUTC, HIDC, nullptr, nullptr);
  gemm_wmma<2><<<gblocks(NNODE, OUTC), 256, 0, stream>>>(
      hc + 1*NH, outW, outb, out + (size_t)NNODE*OUTC,  NNODE, OUTC, HIDC, nullptr, nullptr);
}